// LatticeOrderingLoss_23905787969835
// MI455X (gfx1250) — compile-verified
//
#include <hip/hip_runtime.h>
#include <math.h>

#define BDIM 8192
#define DDIM 256
#define MARGIN 0.1f
#define EPS_F 1e-5f

typedef __attribute__((ext_vector_type(2))) float v2f;
typedef __attribute__((ext_vector_type(8))) float v8f;
typedef __attribute__((ext_vector_type(4))) int   i4;
typedef __attribute__((ext_vector_type(4))) float f4;

// ---------------------------------------------------------------------------
// Kernel 1: d[i] = 2*artanh(min(||x_i||, 1-EPS)) via V_WMMA_F32_16X16X4_F32.
// One wave32 per 16-row tile. A = squared elements (16x4 f32 per ISA layout:
// lanes 0-15 hold row=lane, K=0..1; lanes 16-31 hold row=lane-16, K=2..3),
// B = ones(4x16) so every output column is the row sum. f32 accumulate.
// ---------------------------------------------------------------------------
__global__ __launch_bounds__(256) void norm_d_kernel(const float* __restrict__ X,
                                                     float* __restrict__ d) {
  const int wave = threadIdx.x >> 5;
  const int lane = threadIdx.x & 31;
  const int tile = blockIdx.x * 8 + wave;            // 512 tiles of 16 rows
  const int row  = tile * 16 + (lane & 15);
  const int koff = (lane >> 4) * 2;                  // 0 or 2
  const float* rp = X + row * DDIM + koff;

  v8f c0 = {};
  v8f c1 = {};
  v2f b; b.x = 1.0f; b.y = 1.0f;

  #pragma unroll 8
  for (int k0 = 0; k0 < DDIM; k0 += 8) {
    v2f x0 = *(const v2f*)(rp + k0);
    v2f x1 = *(const v2f*)(rp + k0 + 4);
    v2f a0; a0.x = x0.x * x0.x; a0.y = x0.y * x0.y;
    v2f a1; a1.x = x1.x * x1.x; a1.y = x1.y * x1.y;
    c0 = __builtin_amdgcn_wmma_f32_16x16x4_f32(false, a0, false, b, (short)0, c0,
                                               false, false);
    c1 = __builtin_amdgcn_wmma_f32_16x16x4_f32(false, a1, false, b, (short)0, c1,
                                               false, false);
  }

  // D layout: VGPR v, lanes 0-15 -> M=v (N=lane), lanes 16-31 -> M=v+8.
  if (lane == 0 || lane == 16) {
    const int rbase = tile * 16 + (lane == 16 ? 8 : 0);
    #pragma unroll
    for (int v = 0; v < 8; ++v) {
      float s = c0[v] + c1[v];
      float n = sqrtf(s);
      n = fminf(n, 1.0f - EPS_F);
      d[rbase + v] = 2.0f * atanhf(n);
    }
  }
}

// ---------------------------------------------------------------------------
// Kernel 2: stream ONLY the upper triangle of cmp (halves HBM traffic).
// Block b handles rows (b, B-1-b) -> constant ~B elements per block.
// cmp read with non-temporal b128 loads (read-once, bigger than L2).
// Deterministic block-level reduction -> partials in workspace.
// ---------------------------------------------------------------------------
__global__ __launch_bounds__(256) void pair_kernel(const int* __restrict__ cmp,
                                                   const float* __restrict__ d,
                                                   float* __restrict__ psum,
                                                   unsigned* __restrict__ pcnt) {
  float sum = 0.0f;
  unsigned cnt = 0u;
  const int blk = blockIdx.x;

  #pragma unroll
  for (int rr = 0; rr < 2; ++rr) {
    const int i = (rr == 0) ? blk : (BDIM - 1 - blk);
    const float di = d[i];
    const i4* crow = reinterpret_cast<const i4*>(cmp + (size_t)i * BDIM);
    const int jv0 = (i + 1) >> 2;                    // first vec touching j>i
    for (int jv = jv0 + (int)threadIdx.x; jv < BDIM / 4; jv += 256) {
      i4 c4 = __builtin_nontemporal_load(&crow[jv]);
      f4 d4 = *reinterpret_cast<const f4*>(d + 4 * jv);
      const int j0 = jv * 4;
      #pragma unroll
      for (int t = 0; t < 4; ++t) {
        int   cv  = c4[t];
        bool  act = (cv != 0) && ((j0 + t) > i);
        // cmp==-1 -> relu(+diff+m), cmp==+1 -> relu(-diff+m):  relu(-cv*diff+m)
        float x = fmaf((float)(-cv), di - d4[t], MARGIN);
        sum += act ? fmaxf(x, 0.0f) : 0.0f;
        cnt += act ? 1u : 0u;
      }
    }
  }

  // wave32 reduce (fixed order -> deterministic)
  #pragma unroll
  for (int off = 16; off > 0; off >>= 1) {
    sum += __shfl_xor(sum, off, 32);
    cnt += (unsigned)__shfl_xor((int)cnt, off, 32);
  }
  __shared__ float    wsum[8];
  __shared__ unsigned wcnt[8];
  const int wave = threadIdx.x >> 5;
  const int lane = threadIdx.x & 31;
  if (lane == 0) { wsum[wave] = sum; wcnt[wave] = cnt; }
  __syncthreads();
  if (threadIdx.x == 0) {
    float s = 0.0f; unsigned c = 0u;
    #pragma unroll
    for (int w = 0; w < 8; ++w) { s += wsum[w]; c += wcnt[w]; }
    psum[blk] = s;
    pcnt[blk] = c;
  }
}

// ---------------------------------------------------------------------------
// Kernel 3: deterministic final reduce of BDIM/2 partials + divide.
// ---------------------------------------------------------------------------
__global__ __launch_bounds__(1024) void finish_kernel(const float* __restrict__ psum,
                                                      const unsigned* __restrict__ pcnt,
                                                      float* __restrict__ out) {
  __shared__ float    ss[1024];
  __shared__ unsigned sc[1024];
  float s = 0.0f; unsigned c = 0u;
  for (int idx = threadIdx.x; idx < BDIM / 2; idx += 1024) {
    s += psum[idx];
    c += pcnt[idx];
  }
  ss[threadIdx.x] = s;
  sc[threadIdx.x] = c;
  __syncthreads();
  for (int off = 512; off > 0; off >>= 1) {
    if ((int)threadIdx.x < off) {
      ss[threadIdx.x] += ss[threadIdx.x + off];
      sc[threadIdx.x] += sc[threadIdx.x + off];
    }
    __syncthreads();
  }
  if (threadIdx.x == 0) {
    out[0] = (sc[0] > 0u) ? (ss[0] / (float)sc[0]) : 0.0f;
  }
}

extern "C" void kernel_launch(void* const* d_in, const int* in_sizes, int n_in,
                              void* d_out, int out_size, void* d_ws, size_t ws_size,
                              hipStream_t stream) {
  (void)in_sizes; (void)n_in; (void)out_size; (void)ws_size;
  const float* X   = (const float*)d_in[0];   // [8192, 256] f32
  const int*   cmp = (const int*)d_in[1];     // [8192, 8192] i32
  float* out = (float*)d_out;                 // scalar f32

  // workspace layout: d[8192] f32 | psum[4096] f32 | pcnt[4096] u32
  float*    dvec = (float*)d_ws;
  float*    psum = dvec + BDIM;
  unsigned* pcnt = (unsigned*)(psum + BDIM / 2);

  norm_d_kernel<<<BDIM / 16 / 8, 256, 0, stream>>>(X, dvec);
  pair_kernel<<<BDIM / 2, 256, 0, stream>>>(cmp, dvec, psum, pcnt);
  finish_kernel<<<1, 1024, 0, stream>>>(psum, pcnt, out);
}